// NGFTextureFetch_30502857736195
// MI455X (gfx1250) — compile-verified
//
#include <hip/hip_runtime.h>

// NGF texture fetch: bilinear sampling of [C,16,16,64] f32 maps at 256 random
// points per complex. Memory-bound (~272 MB total traffic -> ~12us @ 23.3TB/s).
// Strategy: stage each 64KB map tile into LDS via CDNA5 async global->LDS
// loads (ASYNCcnt), gather bilinear taps from LDS (ds_load_b128), stream
// results out with NT b128 stores.

#define RES   16
#define FEAT  64
#define PSMP  256   // samples per complex

typedef float f4  __attribute__((ext_vector_type(4)));
typedef int   v4i __attribute__((vector_size(16)));       // matches builtin param

typedef __attribute__((address_space(1))) v4i GlobV4;     // global int4
typedef __attribute__((address_space(3))) v4i LdsV4;      // LDS int4
typedef __attribute__((address_space(3))) float LdsFloat;

__global__ __launch_bounds__(256)
void ngf_texture_fetch_kernel(const float* __restrict__ map,
                              const float* __restrict__ u,
                              const float* __restrict__ v,
                              float* __restrict__ out)
{
    __shared__ __align__(16) float tile[RES * RES * FEAT];   // 64 KB

    const int c   = blockIdx.x;
    const int tid = threadIdx.x;

    // ---------------- Phase 1: async stage map[c] (64 KB) into LDS ----------
    // 256 lanes x 16 B x 16 rounds = 64 KB, layout identical to global.
    const float* gbase = map + (size_t)c * (RES * RES * FEAT);
#pragma unroll
    for (int r = 0; r < 16; ++r) {
        const int idx = r * 1024 + tid * 4;          // float index (16B/lane)
#if __has_builtin(__builtin_amdgcn_global_load_async_to_lds_b128)
        __builtin_amdgcn_global_load_async_to_lds_b128(
            (GlobV4*)(gbase + idx), (LdsV4*)(tile + idx),
            /*imm offset*/0, /*cpol*/0);
#else
        LdsFloat* lp = (LdsFloat*)(tile + idx);
        unsigned lofs = (unsigned)(unsigned long long)lp;   // LDS byte offset
        unsigned long long ga = (unsigned long long)(gbase + idx);
        asm volatile("global_load_async_to_lds_b128 %0, %1, off"
                     :: "v"(lofs), "v"(ga) : "memory");
#endif
    }

#if __has_builtin(__builtin_amdgcn_s_wait_asynccnt)
    __builtin_amdgcn_s_wait_asynccnt(0);
#else
    asm volatile("s_wait_asynccnt 0x0" ::: "memory");
#endif
    __syncthreads();

    // ---------------- Phase 2: bilinear gather from LDS ---------------------
    // 8 threads per sample: thread j of a group handles features [8j, 8j+8)
    // => wave stores 4 samples x 256 B = 1 KB contiguous b128 stores.
    const int j  = tid & 7;       // feature-octet within sample
    const int sg = tid >> 3;      // sample-in-pass, 0..31

    const f4* tl = (const f4*)tile;        // texel t occupies 16 f4's

    for (int pass = 0; pass < 8; ++pass) {
        const int s  = pass * 32 + sg;
        const float uu = u[c * PSMP + s];
        const float vv = v[c * PSMP + s];

        const float x = uu * (float)(RES - 1);
        const float y = vv * (float)(RES - 1);
        int x0 = (int)floorf(x);
        int y0 = (int)floorf(y);
        x0 = x0 < 0 ? 0 : (x0 > RES - 2 ? RES - 2 : x0);
        y0 = y0 < 0 ? 0 : (y0 > RES - 2 ? RES - 2 : y0);
        const float fx = x - (float)x0;
        const float fy = y - (float)y0;
        const float w00 = (1.0f - fx) * (1.0f - fy);
        const float w01 = fx * (1.0f - fy);
        const float w10 = (1.0f - fx) * fy;
        const float w11 = fx * fy;

        // float4 index of (y0,x0) texel's feature block + this thread's octet
        const int b = (y0 * RES + x0) * (FEAT / 4) + 2 * j;
        f4* ov = (f4*)(out + ((size_t)c * PSMP + s) * FEAT + 8 * j);

#pragma unroll
        for (int ch = 0; ch < 2; ++ch) {
            const f4 g00 = tl[b + ch];                              // (y0,  x0)
            const f4 g01 = tl[b + ch + (FEAT / 4)];                 // (y0,  x0+1)
            const f4 g10 = tl[b + ch + RES * (FEAT / 4)];           // (y0+1,x0)
            const f4 g11 = tl[b + ch + (RES + 1) * (FEAT / 4)];     // (y0+1,x0+1)
            f4 o;
            o.x = w00 * g00.x + w01 * g01.x + w10 * g10.x + w11 * g11.x;
            o.y = w00 * g00.y + w01 * g01.y + w10 * g10.y + w11 * g11.y;
            o.z = w00 * g00.z + w01 * g01.z + w10 * g10.z + w11 * g11.z;
            o.w = w00 * g00.w + w01 * g01.w + w10 * g10.w + w11 * g11.w;
            __builtin_nontemporal_store(o, ov + ch);   // write-once stream
        }
    }
}

extern "C" void kernel_launch(void* const* d_in, const int* in_sizes, int n_in,
                              void* d_out, int out_size, void* d_ws, size_t ws_size,
                              hipStream_t stream) {
    const float* map = (const float*)d_in[0];
    const float* u   = (const float*)d_in[1];
    const float* v   = (const float*)d_in[2];
    float* out = (float*)d_out;

    const int C = in_sizes[1] / PSMP;   // number of complexes (2048)

    ngf_texture_fetch_kernel<<<dim3(C), dim3(256), 0, stream>>>(map, u, v, out);
}